// CustomLoss_43860206026829
// MI455X (gfx1250) — compile-verified
//
#include <hip/hip_runtime.h>
#include <stdint.h>

#define W_IMG 4096
#define OUT_DIM 4090            // 4096 - (7-1)
#define TILES_1D 256
#define NTILES (TILES_1D * TILES_1D)

typedef __attribute__((ext_vector_type(2))) float v2f;
typedef __attribute__((ext_vector_type(8))) float v8f;

// ---- CDNA5 WMMA: D(16x16 f32) = A(16x4 f32) x B(4x16 f32) + C ----
__device__ __forceinline__ v8f wmma_k4(v2f a, v2f b, v8f c) {
  // 8-arg pattern: (neg_a, A, neg_b, B, c_mod, C, reuse_a, reuse_b)
  return __builtin_amdgcn_wmma_f32_16x16x4_f32(false, a, false, b, (short)0, c,
                                               false, false);
}

// ---- async global->LDS (CDNA5), per-lane 16B transfers, ASYNCcnt tracked ----
#define ASYNC_LD_B128(ldsaddr32, gptr)                                         \
  asm volatile("global_load_async_to_lds_b128 %0, %1, off" ::"v"(ldsaddr32),   \
               "v"(gptr)                                                       \
               : "memory")

__device__ __forceinline__ void wait_async0() {
#if __has_builtin(__builtin_amdgcn_s_wait_asynccnt)
  __builtin_amdgcn_s_wait_asynccnt(0);
#else
  asm volatile("s_wait_asynccnt 0" ::: "memory");
#endif
}

// ---- order-preserving float <-> uint encoding for atomic min/max ----
__device__ __forceinline__ unsigned encf(float f) {
  unsigned u = __float_as_uint(f);
  return (u & 0x80000000u) ? ~u : (u | 0x80000000u);
}
__device__ __forceinline__ float decf(unsigned u) {
  unsigned v = (u & 0x80000000u) ? (u & 0x7fffffffu) : ~u;
  return __uint_as_float(v);
}

__device__ __forceinline__ float qval(int q, float x, float y) {
  switch (q) {
  case 0: return x;
  case 1: return y;
  case 2: return x * x;
  case 3: return y * y;
  default: return x * y;
  }
}

// ===================== kernel 0: init workspace =====================
__global__ void k_init(unsigned* minmax, double* sums) {
  minmax[0] = 0xffffffffu; // running min (encoded)
  minmax[1] = 0u;          // running max (encoded)
  sums[0] = 0.0;           // mse sum
  sums[1] = 0.0;           // ssim sum
}

// ============ kernel 1: min/max of pred + sum((pred-gt)^2) ============
__global__ __launch_bounds__(256) void k_minmax_mse(
    const float* __restrict__ pred, const float* __restrict__ gt,
    unsigned* __restrict__ minmax, double* __restrict__ sums) {
  const int n4 = (W_IMG * W_IMG) / 4;
  const int tid = blockIdx.x * blockDim.x + threadIdx.x;
  const int stride = gridDim.x * blockDim.x;
  const float4* p4 = (const float4*)pred;
  const float4* g4 = (const float4*)gt;

  float vmin = INFINITY, vmax = -INFINITY, acc = 0.0f;
  for (int i = tid; i < n4; i += stride) {
    __builtin_prefetch((const void*)(p4 + i + stride), 0, 1);
    __builtin_prefetch((const void*)(g4 + i + stride), 0, 1);
    float4 p = p4[i];
    float4 g = g4[i];
    vmin = fminf(vmin, fminf(fminf(p.x, p.y), fminf(p.z, p.w)));
    vmax = fmaxf(vmax, fmaxf(fmaxf(p.x, p.y), fmaxf(p.z, p.w)));
    float dx = p.x - g.x, dy = p.y - g.y, dz = p.z - g.z, dw = p.w - g.w;
    acc = fmaf(dx, dx, acc);
    acc = fmaf(dy, dy, acc);
    acc = fmaf(dz, dz, acc);
    acc = fmaf(dw, dw, acc);
  }
#pragma unroll
  for (int o = 16; o > 0; o >>= 1) {
    vmin = fminf(vmin, __shfl_down(vmin, o));
    vmax = fmaxf(vmax, __shfl_down(vmax, o));
    acc += __shfl_down(acc, o);
  }
  __shared__ float smin[8], smax[8], sacc[8];
  const int wid = threadIdx.x >> 5, lane = threadIdx.x & 31;
  if (lane == 0) { smin[wid] = vmin; smax[wid] = vmax; sacc[wid] = acc; }
  __syncthreads();
  if (threadIdx.x == 0) {
    float m = smin[0], M = smax[0];
    float a = sacc[0];
    for (int i = 1; i < 8; ++i) {
      m = fminf(m, smin[i]);
      M = fmaxf(M, smax[i]);
      a += sacc[i];
    }
    atomicMin(&minmax[0], encf(m));
    atomicMax(&minmax[1], encf(M));
    atomicAdd(&sums[0], (double)a);
  }
}

// ============ kernel 2: SSIM map sum via WMMA banded box filters ============
// One wave per 16x16 output tile. Separable 7x7 window sum:
//   row pass:  Y(16..24 x16) = X(16x24) * Brow(24x16), Brow[k][n]=[n<=k<=n+6]
//   col pass:  Z(16x16)      = Acol(16x24) * Y(24x16),  Acol[m][k]=[m<=k<=m+6]
// Band operand values are identical functions of (lane&15, k) for A and B
// layouts, so one precomputed set serves both passes.
__global__ __launch_bounds__(256) void k_ssim(const float* __restrict__ pred,
                                              const float* __restrict__ gt,
                                              const unsigned* __restrict__ minmax,
                                              double* __restrict__ ssim_sum) {
  __shared__ __align__(16) float s_px[8][24][24];
  __shared__ __align__(16) float s_py[8][24][24];
  __shared__ __align__(16) float s_ys[8][24][16];
  __shared__ float s_blk;

  const int wid = threadIdx.x >> 5;
  const int lane = threadIdx.x & 31;
  if (threadIdx.x == 0) s_blk = 0.0f;
  __syncthreads();

  const float dr = decf(minmax[1]) - decf(minmax[0]);
  const float C1 = (0.01f * dr) * (0.01f * dr);
  const float C2 = (0.03f * dr) * (0.03f * dr);
  const float cn = 49.0f / 48.0f; // sample-covariance normalization
  const float inv49 = 1.0f / 49.0f;

  const int nn = lane & 15;        // N (B/C/D col) or M (A row)
  const int kh = (lane >> 4) << 1; // K sub-position from lane half
  const int half = lane >> 4;

  v2f band[6];
#pragma unroll
  for (int c = 0; c < 6; ++c) {
    int k0 = 4 * c + kh;
    band[c].x = (k0 >= nn && k0 <= nn + 6) ? 1.0f : 0.0f;
    band[c].y = (k0 + 1 >= nn && k0 + 1 <= nn + 6) ? 1.0f : 0.0f;
  }

  float(*px)[24] = s_px[wid];
  float(*py)[24] = s_py[wid];
  float(*ys)[16] = s_ys[wid];

  const int mtop = lane & 15;
  const int mbot0 = 16 + (lane & 15);
  const int mbot = mbot0 > 23 ? 23 : mbot0; // clamp; clamped rows feed unused D rows

  float localsum = 0.0f;
  const int waveGlobal = blockIdx.x * 8 + wid;
  const int nwaves = gridDim.x * 8;

  for (int t = waveGlobal; t < NTILES; t += nwaves) {
    const int tr = t >> 8, tc = t & (TILES_1D - 1);
    const int r0 = tr << 4, c0 = tc << 4;
    const bool interior = (tr < TILES_1D - 1) && (tc < TILES_1D - 1);

    if (interior) {
      // 24 rows x 6 x B128 = 144 lane-transfers per patch, async into LDS
#pragma unroll
      for (int it = 0; it < 5; ++it) {
        int idx = lane + it * 32;
        if (idx < 144) {
          int row = idx / 6, cc = (idx % 6) * 4;
          unsigned lx = (unsigned)(uintptr_t)&px[row][cc];
          unsigned ly = (unsigned)(uintptr_t)&py[row][cc];
          const float* gp = pred + (size_t)(r0 + row) * W_IMG + (c0 + cc);
          const float* gq = gt + (size_t)(r0 + row) * W_IMG + (c0 + cc);
          ASYNC_LD_B128(lx, gp);
          ASYNC_LD_B128(ly, gq);
        }
      }
      wait_async0();
    } else {
      for (int idx = lane; idx < 24 * 24; idx += 32) {
        int row = idx / 24, col = idx % 24;
        int gr = r0 + row, gc = c0 + col;
        bool ok = (gr < W_IMG) && (gc < W_IMG);
        int grc = gr < W_IMG ? gr : (W_IMG - 1);
        int gcc = gc < W_IMG ? gc : (W_IMG - 1);
        size_t off = (size_t)grc * W_IMG + gcc;
        float a = pred[off], b = gt[off];
        px[row][col] = ok ? a : 0.0f;
        py[row][col] = ok ? b : 0.0f;
      }
    }
    __builtin_amdgcn_wave_barrier();

    v8f u[5];
#pragma unroll
    for (int q = 0; q < 5; ++q) {
      v8f ytop = {0, 0, 0, 0, 0, 0, 0, 0};
      v8f ybot = {0, 0, 0, 0, 0, 0, 0, 0};
#pragma unroll
      for (int c = 0; c < 6; ++c) {
        int k0 = 4 * c + kh;
        float xt0 = px[mtop][k0], xt1 = px[mtop][k0 + 1];
        float yt0 = py[mtop][k0], yt1 = py[mtop][k0 + 1];
        float xb0 = px[mbot][k0], xb1 = px[mbot][k0 + 1];
        float yb0 = py[mbot][k0], yb1 = py[mbot][k0 + 1];
        v2f at, ab;
        at.x = qval(q, xt0, yt0);
        at.y = qval(q, xt1, yt1);
        ab.x = qval(q, xb0, yb0);
        ab.y = qval(q, xb1, yb1);
        ytop = wmma_k4(at, band[c], ytop); // rows r0..r0+15 row-filtered
        ybot = wmma_k4(ab, band[c], ybot); // rows r0+16..r0+21 (+pad)
      }
      // stage Y (rows 0..23) into LDS in C/D layout -> re-read as B operands
#pragma unroll
      for (int v = 0; v < 8; ++v) ys[v + half * 8][nn] = ytop[v];
      if (half == 0) {
#pragma unroll
        for (int v = 0; v < 8; ++v) ys[16 + v][nn] = ybot[v];
      }
      __builtin_amdgcn_wave_barrier();

      v8f z = {0, 0, 0, 0, 0, 0, 0, 0};
#pragma unroll
      for (int c = 0; c < 6; ++c) {
        int k0 = 4 * c + kh;
        v2f b;
        b.x = ys[k0][nn];
        b.y = ys[k0 + 1][nn];
        z = wmma_k4(band[c], b, z); // column pass
      }
#pragma unroll
      for (int v = 0; v < 8; ++v) u[q][v] = z[v] * inv49;
      __builtin_amdgcn_wave_barrier(); // ys reused next q
    }

    // SSIM per element; D layout: element (M = v + 8*half, N = nn)
#pragma unroll
    for (int v = 0; v < 8; ++v) {
      int m = v + half * 8;
      int gr = r0 + m, gc = c0 + nn;
      float ux = u[0][v], uy = u[1][v];
      float uxx = u[2][v], uyy = u[3][v], uxy = u[4][v];
      float vx = cn * (uxx - ux * ux);
      float vy = cn * (uyy - uy * uy);
      float vxy = cn * (uxy - ux * uy);
      float S = ((2.0f * ux * uy + C1) * (2.0f * vxy + C2)) /
                ((ux * ux + uy * uy + C1) * (vx + vy + C2));
      if (gr < OUT_DIM && gc < OUT_DIM) localsum += S;
    }
  }

#pragma unroll
  for (int o = 16; o > 0; o >>= 1) localsum += __shfl_down(localsum, o);
  if (lane == 0) atomicAdd(&s_blk, localsum);
  __syncthreads();
  if (threadIdx.x == 0) atomicAdd(ssim_sum, (double)s_blk);
}

// ===================== kernel 3: finalize scalar =====================
__global__ void k_final(const unsigned* __restrict__ minmax,
                        const double* __restrict__ sums,
                        float* __restrict__ out) {
  double mse = sums[0] / ((double)W_IMG * (double)W_IMG);
  double mssim = sums[1] / ((double)OUT_DIM * (double)OUT_DIM);
  out[0] = (float)(mse * exp(1.0 - mssim));
}

extern "C" void kernel_launch(void* const* d_in, const int* in_sizes, int n_in,
                              void* d_out, int out_size, void* d_ws,
                              size_t ws_size, hipStream_t stream) {
  (void)in_sizes; (void)n_in; (void)out_size; (void)ws_size;
  const float* pred = (const float*)d_in[0];
  const float* gt = (const float*)d_in[1];
  unsigned* minmax = (unsigned*)d_ws;
  double* sums = (double*)((char*)d_ws + 8);
  float* out = (float*)d_out;

  k_init<<<1, 1, 0, stream>>>(minmax, sums);
  k_minmax_mse<<<2048, 256, 0, stream>>>(pred, gt, minmax, sums);
  k_ssim<<<2048, 256, 0, stream>>>(pred, gt, minmax, &sums[1]);
  k_final<<<1, 1, 0, stream>>>(minmax, sums, out);
}